// APRConv_36653250904486
// MI455X (gfx1250) — compile-verified
//
#include <hip/hip_runtime.h>
#include <hip/hip_bf16.h>

// Problem constants (from reference)
#define B_ 2
#define CIN_ 32
#define COUT_ 32
#define NPTS 131072
#define KNB 27
#define S_ 3
#define KKTOT 216            // 27 * (32/4) WMMA K-steps of 4
#define WSLOTS (KKTOT*2*32)  // per-(kk,couthalf,lane) weight pairs
#define WLDS_BYTES (WSLOTS*2*4)  // 110,592 B dynamic LDS

typedef float v2f __attribute__((ext_vector_type(2)));
typedef float v8f __attribute__((ext_vector_type(8)));

// ---------------- 1) transpose (B,CIN,N) -> (B,N,CIN) ----------------
__global__ void apr_transpose(const float* __restrict__ in, float* __restrict__ xt) {
    __shared__ float tile[32][33];
    const int b  = blockIdx.y;
    const int n0 = blockIdx.x * 32;
    const int tx = threadIdx.x, ty = threadIdx.y;        // (32,8)
    const float* src = in + (size_t)b * CIN_ * NPTS;
    #pragma unroll
    for (int i = 0; i < 32; i += 8)
        tile[ty + i][tx] = src[(size_t)(ty + i) * NPTS + n0 + tx];
    __syncthreads();
    float* dst = xt + (size_t)b * NPTS * CIN_;
    #pragma unroll
    for (int i = 0; i < 32; i += 8)
        dst[(size_t)(n0 + ty + i) * CIN_ + tx] = tile[tx][ty + i];
}

// ---------------- 2) binning: init / count / offsets / scatter ----------------
__global__ void apr_init(int* __restrict__ perm, int permLen, int* __restrict__ ctr) {
    int i = blockIdx.x * blockDim.x + threadIdx.x;
    if (i < permLen) perm[i] = -1;          // sentinel padding
    if (i < 12) ctr[i] = 0;                 // counts[6] + cursors[6]
}

__device__ __forceinline__ int sidx_of(int lv, int delta) {
    int s = lv + delta;
    return s < 0 ? 0 : (s > S_ - 1 ? S_ - 1 : s);
}

__global__ void apr_count(const int* __restrict__ levels, const int* __restrict__ deltas,
                          int* __restrict__ counts) {
    int n = blockIdx.x * blockDim.x + threadIdx.x;
    if (n >= NPTS) return;
    int lv = levels[n];
    for (int b = 0; b < B_; ++b)
        atomicAdd(&counts[b * S_ + sidx_of(lv, deltas[b])], 1);
}

__global__ void apr_offsets(const int* __restrict__ counts, int* __restrict__ starts,
                            int* __restrict__ padded) {
    if (threadIdx.x == 0 && blockIdx.x == 0) {
        int off = 0;
        for (int i = 0; i < B_ * S_; ++i) {
            int p = (counts[i] + 15) & ~15;   // pad each bin to 16-point tiles
            starts[i] = off; padded[i] = p; off += p;
        }
    }
}

__global__ void apr_scatter(const int* __restrict__ levels, const int* __restrict__ deltas,
                            int* __restrict__ cursors, const int* __restrict__ starts,
                            int* __restrict__ perm) {
    int n = blockIdx.x * blockDim.x + threadIdx.x;
    if (n >= NPTS) return;
    int lv = levels[n];
    for (int b = 0; b < B_; ++b) {
        int bin = b * S_ + sidx_of(lv, deltas[b]);
        int pos = atomicAdd(&cursors[bin], 1);
        perm[starts[bin] + pos] = n;
    }
}

// ---------------- 3) WMMA convolution, 2 point-tiles per wave iteration ----------
// GEMM mapping per tile: D(16 points x 16 couts), x2 cout-halves.
//   A (16x4 f32): M=point on lanes 0-15; VGPR0=K{0|2}, VGPR1=K{1|3} per half-wave
//   B (4x16 f32): K rows striped identically; N=cout on lane&15
//   K-dim order: kk = k*8 + j, channels c = 4*j .. 4*j+3 of neighbor k
// Two tiles share the same B-fragments: 4 WMMAs per 2 ds_load_b64.
__global__ __launch_bounds__(256)
void apr_conv_wmma(const float* __restrict__ xt, const float* __restrict__ weight,
                   const float* __restrict__ bias, const int* __restrict__ nbr,
                   const int* __restrict__ perm, const int* __restrict__ starts,
                   const int* __restrict__ padded, float* __restrict__ out) {
    extern __shared__ float wlds[];                  // WSLOTS pairs, B-fragment order
    const int bs = blockIdx.y;                       // 0..B*S-1
    const int b = bs / S_, s = bs % S_;
    const int tid = threadIdx.x;

    // Stage this s's weight matrix into LDS pre-swizzled into WMMA B-fragments.
    // weight layout: [CIN][S][COUT][27]; c-stride = S*COUT*27
    for (int slot = tid; slot < WSLOTS; slot += 256) {
        int lane = slot & 31;
        int ch   = (slot >> 5) & 1;
        int kk   = slot >> 6;
        int h    = lane >> 4;
        int o    = (lane & 15) + ch * 16;
        int k    = kk >> 3;
        int c    = ((kk & 7) << 2) + 2 * h;
        const float* wp = weight + ((size_t)c * S_ + s) * (COUT_ * KNB) + o * KNB + k;
        wlds[slot * 2 + 0] = wp[0];
        wlds[slot * 2 + 1] = wp[(size_t)S_ * COUT_ * KNB];   // c+1
    }
    __syncthreads();

    const int nTiles   = padded[bs] >> 4;
    const int binStart = starts[bs];
    const int lane = tid & 31;
    const int wave = tid >> 5;
    const int h    = lane >> 4;
    const int ln15 = lane & 15;
    const float bias0 = bias[ln15];
    const float bias1 = bias[ln15 + 16];
    const float* xtb  = xt + (size_t)b * NPTS * CIN_;
    const float* wlp  = wlds + lane * 2;             // per-lane LDS base for B frags

    const int nPairs     = (nTiles + 1) >> 1;
    const int waveGlobal = blockIdx.x * 8 + wave;
    const int waveStride = gridDim.x * 8;

    for (int tp = waveGlobal; tp < nPairs; tp += waveStride) {
        const int t0 = tp * 2, t1 = tp * 2 + 1;
        int pidA = perm[binStart + (t0 << 4) + ln15];
        int pidB = (t1 < nTiles) ? perm[binStart + (t1 << 4) + ln15] : -1;
        int pidA0 = pidA < 0 ? 0 : pidA;
        int pidB0 = pidB < 0 ? 0 : pidB;
        v8f accA0 = {}, accA1 = {}, accB0 = {}, accB1 = {};
        const int* nrowA = nbr + (size_t)pidA0 * KNB;
        const int* nrowB = nbr + (size_t)pidB0 * KNB;

        for (int k = 0; k < KNB; ++k) {
            const float* rowA = xtb + ((size_t)nrowA[k] << 5) + 2 * h;  // 128B row
            const float* rowB = xtb + ((size_t)nrowB[k] << 5) + 2 * h;
            const float* wk   = wlp + (size_t)k * 8 * 128;              // k-step LDS base
            #pragma unroll
            for (int j = 0; j < 8; ++j) {
                v2f a0 = *(const v2f*)(rowA + 4 * j);               // global b64 gather
                v2f a1 = *(const v2f*)(rowB + 4 * j);
                v2f b0 = *(const v2f*)(wk + j * 128);               // ds b64 (couts 0-15)
                v2f b1 = *(const v2f*)(wk + j * 128 + 64);          // ds b64 (couts 16-31)
                accA0 = __builtin_amdgcn_wmma_f32_16x16x4_f32(
                            false, a0, false, b0, (short)0, accA0, false, false);
                accB0 = __builtin_amdgcn_wmma_f32_16x16x4_f32(
                            false, a1, false, b0, (short)0, accB0, false, false);
                accA1 = __builtin_amdgcn_wmma_f32_16x16x4_f32(
                            false, a0, false, b1, (short)0, accA1, false, false);
                accB1 = __builtin_amdgcn_wmma_f32_16x16x4_f32(
                            false, a1, false, b1, (short)0, accB1, false, false);
            }
        }

        // D layout: VGPR v -> row M = v (lanes 0-15) / v+8 (lanes 16-31); N = ln15
        float* o0 = out + (size_t)(b * COUT_ + ln15) * NPTS;
        float* o1 = out + (size_t)(b * COUT_ + ln15 + 16) * NPTS;
        #pragma unroll
        for (int v = 0; v < 8; ++v) {
            int pvA = __shfl(pidA, v + 8 * h, 32);
            if (pvA >= 0) {
                o0[pvA] = accA0[v] + bias0;
                o1[pvA] = accA1[v] + bias1;
            }
        }
        #pragma unroll
        for (int v = 0; v < 8; ++v) {
            int pvB = __shfl(pidB, v + 8 * h, 32);
            if (pvB >= 0) {
                o0[pvB] = accB0[v] + bias0;
                o1[pvB] = accB1[v] + bias1;
            }
        }
    }
}

// ---------------- launch ----------------
extern "C" void kernel_launch(void* const* d_in, const int* in_sizes, int n_in,
                              void* d_out, int out_size, void* d_ws, size_t ws_size,
                              hipStream_t stream) {
    const float* inF    = (const float*)d_in[0];
    const float* weight = (const float*)d_in[1];
    const float* bias   = (const float*)d_in[2];
    const int*   nbr    = (const int*)d_in[3];
    const int*   levels = (const int*)d_in[4];
    const int*   deltas = (const int*)d_in[5];
    float*       out    = (float*)d_out;

    // Workspace layout: xt (33.5 MB) | perm | counts/cursors/starts/padded
    float* xt = (float*)d_ws;
    const int permLen = B_ * NPTS + B_ * S_ * 16;
    int* perm    = (int*)(xt + (size_t)B_ * NPTS * CIN_);
    int* counts  = perm + permLen;
    int* cursors = counts + 6;
    int* starts  = counts + 12;
    int* padded  = counts + 18;

    apr_transpose<<<dim3(NPTS / 32, B_), dim3(32, 8), 0, stream>>>(inF, xt);
    apr_init<<<(permLen + 255) / 256, 256, 0, stream>>>(perm, permLen, counts);
    apr_count<<<NPTS / 256, 256, 0, stream>>>(levels, deltas, counts);
    apr_offsets<<<1, 32, 0, stream>>>(counts, starts, padded);
    apr_scatter<<<NPTS / 256, 256, 0, stream>>>(levels, deltas, cursors, starts, perm);
    apr_conv_wmma<<<dim3(128, B_ * S_), 256, WLDS_BYTES, stream>>>(
        xt, weight, bias, nbr, perm, starts, padded, out);
}